// IAFM_24996709663326
// MI455X (gfx1250) — compile-verified
//
#include <hip/hip_runtime.h>
#include <hip/hip_bf16.h>

// Pairwise-interaction factorization machine scoring:
//   d[i]  = dot(vecs[f0[i]], vecs[f1[i]])            (D = 64, f32)
//   out[seg[i]] += W[intr[i]]/div[i] * d[i] + b
//
// MI455X strategy:
//  * Gather-bound; vecs table (25.6 MB) is L2-resident (192 MB L2).
//  * 16 terms per wave-group; dots via V_WMMA_F32_16X16X4_F32 (f32 precision,
//    FLOPs are free at 0.25 FLOP/B intensity). Diagonal of the 16x16 product
//    is the 16 dot products.
//  * K-permuted loading: since WMMA just sums over K, chunk pair (2c, 2c+1)
//    uses k = 8c + 4*kh + j, so each lane fetches one float4 (b128) feeding
//    two WMMA chunks -> 16 b128 gathers per lane-group instead of 32 b64.
//  * Segment reduction via global_atomic_add_f32 (segment ids sorted,
//    ~256 terms/segment, output is L2-resident).

typedef __attribute__((ext_vector_type(2))) float v2f;
typedef __attribute__((ext_vector_type(4))) float v4f;
typedef __attribute__((ext_vector_type(8))) float v8f;

#define DDIM 64

__global__ void iafm_zero_kernel(float* __restrict__ out, int n) {
    int i = blockIdx.x * blockDim.x + threadIdx.x;
    if (i < n) out[i] = 0.0f;
}

__global__ __launch_bounds__(256) void iafm_wmma_kernel(
    const int*   __restrict__ feat_idxs,    // [N,2]
    const int*   __restrict__ intr_idxs,    // [N]
    const float* __restrict__ intr_divs,    // [N]
    const int*   __restrict__ segment_ids,  // [N]
    const float* __restrict__ vecs,         // [N_FEATS, 64]
    const float* __restrict__ intr_W,       // [N_INTRS, 1]
    const float* __restrict__ intr_b,       // [1]
    float*       __restrict__ out,          // [B]
    int n_terms)
{
    const int lane            = threadIdx.x & 31;
    const int wave_in_block   = threadIdx.x >> 5;
    const int waves_per_block = blockDim.x >> 5;
    const int global_wave     = blockIdx.x * waves_per_block + wave_in_block;
    const int n_waves         = gridDim.x * waves_per_block;
    const int n_groups        = n_terms >> 4;     // 16 terms per group

    // A (16x4 f32, 2 VGPRs): lane t holds row m = t&15, K-slot pair of
    // half kh = t>>4.  B (4x16) is the symmetric layout with n = t&15, so
    // each lane gathers the SAME (term, k) slice from v0 (A) and v1 (B).
    const int m  = lane & 15;
    const int kh = lane >> 4;

    const float bias = intr_b[0];

    for (int g = global_wave; g < n_groups; g += n_waves) {
        const int term = (g << 4) + m;
        const int2 fp  = *(const int2*)(feat_idxs + (size_t)term * 2);
        // This lane's 16B-aligned slice base: k = 4*kh within each 8-float
        // super-chunk.
        const float* r0 = vecs + (size_t)fp.x * DDIM + 4 * kh;
        const float* r1 = vecs + (size_t)fp.y * DDIM + 4 * kh;

        // Prefetch next group's index stream (global_prefetch_b8).
        {
            const int gn = g + n_waves;
            if (gn < n_groups)
                __builtin_prefetch(feat_idxs + (((size_t)(gn << 4) + m) * 2), 0, 1);
        }

        v8f acc = {};
        #pragma unroll
        for (int c = 0; c < 8; ++c) {           // 8 super-chunks of K=8
            v4f q0 = *(const v4f*)(r0 + 8 * c); // k = 8c + 4kh + {0..3} of v0
            v4f q1 = *(const v4f*)(r1 + 8 * c); // same k slice of v1
            v2f a0 = __builtin_shufflevector(q0, q0, 0, 1);
            v2f b0 = __builtin_shufflevector(q1, q1, 0, 1);
            v2f a1 = __builtin_shufflevector(q0, q0, 2, 3);
            v2f b1 = __builtin_shufflevector(q1, q1, 2, 3);
            // 8 args: (neg_a, A, neg_b, B, c_mod, C, reuse_a, reuse_b)
            acc = __builtin_amdgcn_wmma_f32_16x16x4_f32(
                false, a0, false, b0, (short)0, acc, false, false);
            acc = __builtin_amdgcn_wmma_f32_16x16x4_f32(
                false, a1, false, b1, (short)0, acc, false, false);
        }

        // Diagonal extraction (C/D layout):
        //   md in 0..7  -> lane md,     VGPR md
        //   md in 8..15 -> lane md+16,  VGPR md-8
        const bool has = (lane < 8) || (lane >= 24);
        if (has) {
            const int md = (lane < 8) ? lane : (lane - 16);
            const int vg = ((lane < 8) ? lane : (lane - 24)) & 7;
            // Branchless select chain -> v_cndmask ladder.
            float d = acc[0];
            d = (vg == 1) ? acc[1] : d;
            d = (vg == 2) ? acc[2] : d;
            d = (vg == 3) ? acc[3] : d;
            d = (vg == 4) ? acc[4] : d;
            d = (vg == 5) ? acc[5] : d;
            d = (vg == 6) ? acc[6] : d;
            d = (vg == 7) ? acc[7] : d;

            const int   termd = (g << 4) + md;
            const int   seg   = segment_ids[termd];
            const float w     = intr_W[intr_idxs[termd]];
            const float dv    = intr_divs[termd];
            const float val   = w / dv * d + bias;
            atomicAdd(out + seg, val);          // global_atomic_add_f32
        }
    }

    // Scalar tail for n_terms not divisible by 16 (not hit for N = 2^21).
    const int tail_base = n_groups << 4;
    const int tail_cnt  = n_terms - tail_base;
    if (tail_cnt > 0 && global_wave == 0 && lane < tail_cnt) {
        const int term = tail_base + lane;
        const int2 fp  = *(const int2*)(feat_idxs + (size_t)term * 2);
        const float* r0 = vecs + (size_t)fp.x * DDIM;
        const float* r1 = vecs + (size_t)fp.y * DDIM;
        float d = 0.0f;
        #pragma unroll 8
        for (int k = 0; k < DDIM; ++k) d += r0[k] * r1[k];
        const float w  = intr_W[intr_idxs[term]];
        const float dv = intr_divs[term];
        atomicAdd(out + segment_ids[term], w / dv * d + bias);
    }
}

extern "C" void kernel_launch(void* const* d_in, const int* in_sizes, int n_in,
                              void* d_out, int out_size, void* d_ws, size_t ws_size,
                              hipStream_t stream) {
    (void)n_in; (void)d_ws; (void)ws_size;
    const int*   feat_idxs   = (const int*)d_in[0];
    const int*   intr_idxs   = (const int*)d_in[1];
    const float* intr_divs   = (const float*)d_in[2];
    const int*   segment_ids = (const int*)d_in[3];
    const float* vecs        = (const float*)d_in[4];
    const float* intr_W      = (const float*)d_in[5];
    const float* intr_b      = (const float*)d_in[6];
    float*       out         = (float*)d_out;

    const int n_terms = in_sizes[1];   // intr_idxs element count == N

    // Zero the (poisoned) output accumulator.
    iafm_zero_kernel<<<(out_size + 255) / 256, 256, 0, stream>>>(out, out_size);

    // 256 threads = 8 wave32s per block; one group of 16 terms per wave-iter.
    const int waves_per_block = 256 / 32;
    const int n_groups = n_terms >> 4;
    int blocks = (n_groups + waves_per_block - 1) / waves_per_block;
    if (blocks > 16384) blocks = 16384;
    if (blocks < 1)     blocks = 1;

    iafm_wmma_kernel<<<blocks, 256, 0, stream>>>(
        feat_idxs, intr_idxs, intr_divs, segment_ids,
        vecs, intr_W, intr_b, out, n_terms);
}